// QueryAndGroup_32246614459287
// MI455X (gfx1250) — compile-verified
//
#include <hip/hip_runtime.h>

// ---------------------------------------------------------------------------
// Ball query + grouping (pointnet2 semantics) for MI455X / gfx1250, wave32.
// Phase 1: distance tiles via V_WMMA_F32_16X16X4_F32, homogeneous coords:
//   A[m] = (-2qx, -2qy, -2qz, 1),  B[n] = (px, py, pz, |p|^2)
//   => D[m][n] = -2 q.p + |p|^2 ;  hit iff D + |q|^2 < r^2
// Phase 2: feature transpose (B,C,N)->(B,N,C) in scratch, then gather with
//   lane = sample slot -> contiguous 256B per sample reads, coalesced stores.
// ---------------------------------------------------------------------------

typedef float v2f __attribute__((ext_vector_type(2)));
typedef float v8f __attribute__((ext_vector_type(8)));

#define BQ    4
#define NQPT  2048
#define NPTS  16384
#define NCH   64
#define NS    32
#define R2    0.01f           // radius^2 = 0.1^2

__device__ __forceinline__ unsigned ballot32(bool p) {
#if __has_builtin(__builtin_amdgcn_ballot_w32)
  return __builtin_amdgcn_ballot_w32(p);
#else
  return (unsigned)__ballot(p);
#endif
}

__device__ __forceinline__ float readlane_f(float v, int l) {
  return __int_as_float(__builtin_amdgcn_readlane(__float_as_int(v), l));
}

// ---------------------------------------------------------------------------
// Kernel 1: ball query. One wave handles 16 queries; scans support points in
// chunks of 16 via one f32 16x16x4 WMMA per chunk, with a 1-chunk software
// prefetch pipeline so the b96 point load overlaps WMMA + mask processing.
// ---------------------------------------------------------------------------
__global__ __launch_bounds__(256) void bq_wmma_kernel(
    const float* __restrict__ query,    // (B, NQPT, 3)
    const float* __restrict__ support,  // (B, NPTS, 3)
    int* __restrict__ idx_out)          // (B*NQPT, NS)
{
  __shared__ int lds_idx[8 * 16 * NS];
  __shared__ int lds_cnt[8 * 16];

  const int lane = threadIdx.x & 31;
  const int widx = threadIdx.x >> 5;
  const int gw   = blockIdx.x * 8 + widx;
  const int q0   = gw * 16;                 // first flat query (b*NQPT + q)
  const int b    = q0 / NQPT;
  const int m    = lane & 15;
  const bool lo  = (lane < 16);

  // ---- A matrix: (-2qx,-2qy) lower half, (-2qz, 1) upper -----------------
  const float* qb = query + (size_t)q0 * 3;
  const float qx = qb[m * 3 + 0], qy = qb[m * 3 + 1], qz = qb[m * 3 + 2];
  const float qq = qx * qx + qy * qy + qz * qz;
  v2f A;
  A.x = -2.0f * (lo ? qx : qz);
  A.y = lo ? (-2.0f * qy) : 1.0f;

  // qv[v]: |q|^2 of row v (lanes 0-15) / row v+8 (lanes 16-31)
  float qv[8];
#pragma unroll
  for (int v = 0; v < 8; ++v) {
    const float a = readlane_f(qq, v);
    const float c = readlane_f(qq, v + 8);
    qv[v] = lo ? a : c;
  }

  int cnt[16];
#pragma unroll
  for (int i = 0; i < 16; ++i) cnt[i] = 0;

  const float* sb = support + (size_t)b * NPTS * 3;

  // prime the pipeline: chunk 0 point
  float px, py, pz;
  {
    const float* p = sb + (size_t)m * 3;
    px = p[0]; py = p[1]; pz = p[2];
  }

  for (int n0 = 0; n0 < NPTS; n0 += 16) {
    const float cpx = px, cpy = py, cpz = pz;
    // prefetch next chunk (clamped, uniform -> no divergence, no branch)
    {
      const int nn = (n0 + 16 < NPTS) ? (n0 + 16) : (NPTS - 16);
      const float* pn = sb + (size_t)(nn + m) * 3;
      px = pn[0]; py = pn[1]; pz = pn[2];
    }

    const float pp = cpx * cpx + cpy * cpy + cpz * cpz;   // |p|^2
    v2f Bm;
    Bm.x = lo ? cpx : cpz;                                // K0 / K2
    Bm.y = lo ? cpy : pp;                                 // K1 / K3

    v8f Cz = {0.f, 0.f, 0.f, 0.f, 0.f, 0.f, 0.f, 0.f};
    v8f D = __builtin_amdgcn_wmma_f32_16x16x4_f32(
        false, A, false, Bm, (short)0, Cz, false, false);

#pragma unroll
    for (int v = 0; v < 8; ++v) {
      const bool hit = (D[v] + qv[v]) < R2;
      const unsigned msk = ballot32(hit);       // lo16: row v, hi16: row v+8
      if (msk == 0u) continue;
      const unsigned mlo = msk & 0xFFFFu;
      const unsigned mhi = msk >> 16;
      const unsigned pm = lo ? (msk & 0xFFFFu) : (msk & 0xFFFF0000u);
      const int prefix  = (int)__builtin_amdgcn_mbcnt_lo(pm, 0u);
      const int base    = lo ? cnt[v] : cnt[v + 8];
      const int slot    = base + prefix;
      const int row     = lo ? v : (v + 8);
      if (hit && slot < NS)
        lds_idx[(widx * 16 + row) * NS + slot] = n0 + m;
      cnt[v]     = min(cnt[v]     + (int)__popc(mlo), NS);
      cnt[v + 8] = min(cnt[v + 8] + (int)__popc(mhi), NS);
    }

    int mn = cnt[0];
#pragma unroll
    for (int i = 1; i < 16; ++i) mn = min(mn, cnt[i]);
    if (mn >= NS) break;
  }

  int cm = 0;
#pragma unroll
  for (int i = 0; i < 16; ++i) cm = (m == i) ? cnt[i] : cm;
  if (lo) lds_cnt[widx * 16 + lane] = cm;
  __syncthreads();

  for (int mm = 0; mm < 16; ++mm) {
    const int c     = lds_cnt[widx * 16 + mm];
    const int first = (c > 0) ? lds_idx[(widx * 16 + mm) * NS] : 0;
    const int vi    = (lane < c) ? lds_idx[(widx * 16 + mm) * NS + lane] : first;
    idx_out[(size_t)(q0 + mm) * NS + lane] = vi;
  }
}

// ---------------------------------------------------------------------------
// Feature transpose: (B, NCH, NPTS) -> (B, NPTS, NCH), classic 32x32 LDS tile.
// Reads coalesced along n, writes coalesced along c.
// ---------------------------------------------------------------------------
__global__ __launch_bounds__(256) void transpose_kernel(
    const float* __restrict__ f, float* __restrict__ ft)
{
  __shared__ float tile[32][33];
  const int tx = threadIdx.x, ty = threadIdx.y;
  const int n0 = blockIdx.x * 32;
  const int b  = blockIdx.y >> 1;
  const int c0 = (blockIdx.y & 1) * 32;

  const float* fb = f + (size_t)b * NCH * NPTS;
#pragma unroll
  for (int j = 0; j < 32; j += 8)
    tile[ty + j][tx] = fb[(size_t)(c0 + ty + j) * NPTS + (n0 + tx)];
  __syncthreads();

  float* ob = ft + (size_t)b * NPTS * NCH;
#pragma unroll
  for (int j = 0; j < 32; j += 8)
    ob[(size_t)(n0 + ty + j) * NCH + (c0 + tx)] = tile[tx][ty + j];
}

// ---------------------------------------------------------------------------
// Kernel 2 (fast path): gather from transposed features.
// One wave per query, lane = sample slot. Per sample, all 64 channels live in
// 256B contiguous -> 2 L2 lines, re-hit from L0 across the c-loop.
// ---------------------------------------------------------------------------
template <bool TRANSPOSED>
__global__ __launch_bounds__(256) void gather_kernel(
    const float* __restrict__ query,    // (B, NQPT, 3)
    const float* __restrict__ support,  // (B, NPTS, 3)
    const float* __restrict__ feats,    // TRANSPOSED ? (B,NPTS,NCH) : (B,NCH,NPTS)
    const int*   __restrict__ idx,      // (B*NQPT, NS)
    float* __restrict__ out)            // [xyz (B,3,NQPT,NS)] ++ [feat (B,NCH,NQPT,NS)]
{
  const int lane = threadIdx.x & 31;
  const int g    = (int)((blockIdx.x * blockDim.x + threadIdx.x) >> 5);
  const int b    = g / NQPT;
  const int q    = g - b * NQPT;

  const int myidx = idx[(size_t)g * NS + lane];

  const float* sp = support + ((size_t)b * NPTS + myidx) * 3;
  const float sx = sp[0], sy = sp[1], sz = sp[2];
  const float* qp = query + (size_t)g * 3;          // uniform -> scalar loads
  const float cx = qp[0], cy = qp[1], cz = qp[2];

  out[(((size_t)b * 3 + 0) * NQPT + q) * NS + lane] = sx - cx;
  out[(((size_t)b * 3 + 1) * NQPT + q) * NS + lane] = sy - cy;
  out[(((size_t)b * 3 + 2) * NQPT + q) * NS + lane] = sz - cz;

  const size_t OFF = (size_t)BQ * 3 * NQPT * NS;    // 786432
  float* ob = out + OFF;

  if (TRANSPOSED) {
    const float* fT = feats + ((size_t)b * NPTS + myidx) * NCH;   // 256B row
#pragma unroll 8
    for (int c = 0; c < NCH; ++c)
      ob[(((size_t)b * NCH + c) * NQPT + q) * NS + lane] = fT[c];
  } else {
    const float* fb = feats + (size_t)b * NCH * NPTS;
#pragma unroll 4
    for (int c = 0; c < NCH; ++c)
      ob[(((size_t)b * NCH + c) * NQPT + q) * NS + lane] =
          fb[(size_t)c * NPTS + myidx];
  }
}

// ---------------------------------------------------------------------------
extern "C" void kernel_launch(void* const* d_in, const int* in_sizes, int n_in,
                              void* d_out, int out_size, void* d_ws, size_t ws_size,
                              hipStream_t stream) {
  const float* query   = (const float*)d_in[0];   // (4,2048,3)
  const float* support = (const float*)d_in[1];   // (4,16384,3)
  const float* feats   = (const float*)d_in[2];   // (4,64,16384)
  float*       out     = (float*)d_out;

  int* ws_idx = (int*)d_ws;                                   // 1 MB
  const size_t IDX_BYTES = (size_t)BQ * NQPT * NS * sizeof(int);
  const size_t FT_BYTES  = (size_t)BQ * NPTS * NCH * sizeof(float);  // 16 MB

  // Phase 1: 8192 queries / 16 per wave / 8 waves per block = 64 blocks
  bq_wmma_kernel<<<(BQ * NQPT) / (16 * 8), 256, 0, stream>>>(query, support, ws_idx);

  if (ws_size >= IDX_BYTES + FT_BYTES) {
    // Fast path: transpose features into scratch, then dense-row gather.
    float* featT = (float*)((char*)d_ws + IDX_BYTES);
    dim3 tb(32, 8, 1);
    dim3 tg(NPTS / 32, (NCH / 32) * BQ, 1);
    transpose_kernel<<<tg, tb, 0, stream>>>(feats, featT);
    gather_kernel<true><<<(BQ * NQPT) / 8, 256, 0, stream>>>(
        query, support, featT, ws_idx, out);
  } else {
    gather_kernel<false><<<(BQ * NQPT) / 8, 256, 0, stream>>>(
        query, support, feats, ws_idx, out);
  }
}